// SoftMaxLoss_27470610825694
// MI455X (gfx1250) — compile-verified
//
#include <hip/hip_runtime.h>

// ---------------------------------------------------------------------------
// SoftMaxLoss (gather + -log + mean) for MI455X / gfx1250, wave32.
//
// Latency-bound problem (~100 KB useful traffic vs 23.3 TB/s HBM), so:
//   * single 1024-thread workgroup (32 waves), one row-gather per thread-iter
//   * deterministic two-level reduction (no float atomics)
//   * intra-wave 32-lane sum done with V_WMMA_F32_16X16X4_F32 against a
//     ones B-matrix: D[m][n] = partial[m] + partial[m+16]  (exact f32 FMA)
// ---------------------------------------------------------------------------

typedef __attribute__((ext_vector_type(2))) float v2f;
typedef __attribute__((ext_vector_type(8))) float v8f;

__device__ __forceinline__ float readlane_f(float v, int lane) {
    return __builtin_bit_cast(float,
        __builtin_amdgcn_readlane(__builtin_bit_cast(int, v), lane));
}

// Sum of one float per lane across the full wave32; EXEC must be all ones.
// Uses the 16x4 f32 A-layout: lane L<16 holds A[L][0],A[L][1]; lane 16+L
// holds A[L][2],A[L][3]. With a = {partial, 0} and B = ones:
//   D[m][n] = partial[m] + partial[m+16]   for every n.
// C/D layout: lane n<16, VGPR v -> D[v][n]; lane n>=16 -> D[v+8][n-16].
// So sum(d[0..7]) on lane0 = sum_{m=0..7}(p_m+p_{m+16}), lane16 holds the
// complementary half; readlane(0)+readlane(16) = full 32-lane sum.
__device__ __forceinline__ float wave_sum_wmma(float x) {
    v2f a; a.x = x;    a.y = 0.0f;
    v2f b; b.x = 1.0f; b.y = 1.0f;
    v8f c = {};
    v8f d = __builtin_amdgcn_wmma_f32_16x16x4_f32(
        /*neg_a=*/false, a, /*neg_b=*/false, b,
        /*c_mod=*/(short)0, c, /*reuse_a=*/false, /*reuse_b=*/false);
    float s = d[0] + d[1] + d[2] + d[3] + d[4] + d[5] + d[6] + d[7];
    return readlane_f(s, 0) + readlane_f(s, 16);
}

__global__ __launch_bounds__(1024) void SoftMaxLoss_27470610825694_kernel(
    const float* __restrict__ output,   // [B, C] f32
    const int*   __restrict__ label,    // [B]    (harness contract: int32)
    float*       __restrict__ res,      // [1]
    int B, long long C) {
    const int tid = threadIdx.x;
    const float NEG_LOG_EPS = 13.815510557964274f;  // -log(1e-6)

    float acc = 0.0f;
    for (int i = tid; i < B; i += 1024) {
        int   lbl = label[i];
        float p   = output[(long long)i * C + (long long)lbl];  // >2GiB: 64-bit addr
        float lp  = logf(p);
        acc += (lp != 0.0f) ? -lp : NEG_LOG_EPS;   // -0.0 -> EPS branch, matches ref
    }

    // Level 1: per-wave sum via WMMA (all 32 waves fully active -> EXEC all 1s)
    float wsum = wave_sum_wmma(acc);

    __shared__ float lds[32];
    const int wave = tid >> 5;
    const int lane = tid & 31;
    if (lane == 0) lds[wave] = wsum;
    __syncthreads();

    // Level 2: wave 0 reduces the 32 per-wave partials (its EXEC is all 1s)
    if (wave == 0) {
        float t = wave_sum_wmma(lds[lane]);
        if (lane == 0) res[0] = t / (float)B;
    }
}

extern "C" void kernel_launch(void* const* d_in, const int* in_sizes, int n_in,
                              void* d_out, int out_size, void* d_ws, size_t ws_size,
                              hipStream_t stream) {
    const float* output = (const float*)d_in[0];
    const int*   label  = (const int*)d_in[1];
    const int       B = in_sizes[1];                       // 8192
    const long long C = (long long)in_sizes[0] / (long long)B;  // 32000

    SoftMaxLoss_27470610825694_kernel<<<1, 1024, 0, stream>>>(
        output, label, (float*)d_out, B, C);
}